// HGTLayer_22978075033752
// MI455X (gfx1250) — compile-verified
//
#include <hip/hip_runtime.h>
#include <hip/hip_bf16.h>
#include <cmath>

// ---------------- problem constants (match reference) ----------------
#define NA      40000
#define NB      40000
#define NEDGE   200000
#define IN_DIM  128
#define OUT_DIM 128
#define NH      8
#define DKH     16      // OUT/H ; 1/sqrt(DK) = 0.25

typedef float v2f __attribute__((ext_vector_type(2)));
typedef float v8f __attribute__((ext_vector_type(8)));

// ---------------- utility fills ----------------
__global__ void fill_kernel(float* __restrict__ p, float v, int n) {
    int i = blockIdx.x * blockDim.x + threadIdx.x;
    if (i < n) p[i] = v;
}

// CAS-based float atomic max (handles negatives; -inf init)
__device__ inline void atomicMaxF(float* addr, float val) {
    int* ia = (int*)addr;
    int cur = *ia;
    while (__int_as_float(cur) < val) {
        int prev = atomicCAS(ia, cur, __float_as_int(val));
        if (prev == cur) break;
        cur = prev;
    }
}

__device__ inline void load16(const float* __restrict__ p, float* r) {
    const float4* p4 = (const float4*)p;          // 64B-aligned (head offsets)
    float4 x0 = p4[0], x1 = p4[1], x2 = p4[2], x3 = p4[3];
    r[0]=x0.x; r[1]=x0.y; r[2]=x0.z;  r[3]=x0.w;
    r[4]=x1.x; r[5]=x1.y; r[6]=x1.z;  r[7]=x1.w;
    r[8]=x2.x; r[9]=x2.y; r[10]=x2.z; r[11]=x2.w;
    r[12]=x3.x;r[13]=x3.y;r[14]=x3.z; r[15]=x3.w;
}

// ---------------- fused K/Q/V projection (WMMA fp32) ----------------
// 256 threads = 8 waves; each wave owns a 16-row slab. A-row fragments are
// loaded ONCE into registers (32 x v2f = 64 VGPRs) and reused across the
// 3 weight matrices x 8 N-tiles. Weight N-tile staged in LDS transposed.
__global__ __launch_bounds__(256)
void gemm_kqv_kernel(const float* __restrict__ A,
                     const float* __restrict__ Wk, const float* __restrict__ Wq,
                     const float* __restrict__ Wv,
                     const float* __restrict__ bk, const float* __restrict__ bq,
                     const float* __restrict__ bv,
                     float* __restrict__ K, float* __restrict__ Q, float* __restrict__ V,
                     int M)
{
    __shared__ float Wl[16][130];   // [n][k], padded vs bank conflicts
    const int lane = threadIdx.x & 31;
    const int wave = threadIdx.x >> 5;
    const int half = lane >> 4;
    const int nrow = lane & 15;     // m-row for A frag, n-col for B/C frags
    const int m0   = blockIdx.x * 128 + wave * 16;
    const bool active = (m0 < M);
    const float* Arow = A + (size_t)(m0 + nrow) * IN_DIM;

    // A fragments: af[i] covers K = 4*i + {0,1} + 2*half  (ISA 16x4 A layout)
    v2f af[32];
    #pragma unroll
    for (int i = 0; i < 32; ++i) {
        if (active) af[i] = *(const v2f*)(Arow + 4 * i + 2 * half);
        else { af[i][0] = 0.0f; af[i][1] = 0.0f; }
    }

    const float* Ws[3] = { Wk, Wq, Wv };
    const float* Bs[3] = { bk, bq, bv };
    float*       Ds[3] = { K,  Q,  V  };

    for (int w = 0; w < 3; ++w) {
        const float* Wc = Ws[w];
        const float* bc = Bs[w];
        float*       Dc = Ds[w];
        for (int nt = 0; nt < 8; ++nt) {
            const int n0 = nt * 16;
            __syncthreads();
            for (int idx = threadIdx.x; idx < 128 * 16; idx += 256) {
                int k = idx >> 4, n = idx & 15;
                Wl[n][k] = Wc[k * OUT_DIM + n0 + n];
            }
            __syncthreads();
            if (!active) continue;

            const float bval = bc[n0 + nrow];
            v8f acc;
            #pragma unroll
            for (int j = 0; j < 8; ++j) acc[j] = bval;

            #pragma unroll
            for (int i = 0; i < 32; ++i) {
                v2f b;
                b[0] = Wl[nrow][4 * i + 2 * half];
                b[1] = Wl[nrow][4 * i + 2 * half + 1];
                acc = __builtin_amdgcn_wmma_f32_16x16x4_f32(
                    false, af[i], false, b, (short)0, acc, false, false);
            }
            #pragma unroll
            for (int j = 0; j < 8; ++j)
                Dc[(size_t)(m0 + j + 8 * half) * OUT_DIM + n0 + nrow] = acc[j];
        }
    }
}

// ---------------- output GEMM with fused HGT epilogue ----------------
// trans = T @ Wa + ba ; out = (has_msg ? trans : feat)*alpha + feat*(1-alpha)
__global__ __launch_bounds__(256)
void gemm_out_kernel(const float* __restrict__ T, const float* __restrict__ W,
                     const float* __restrict__ bias, const float* __restrict__ feat,
                     const float* __restrict__ denom, const float* __restrict__ skip,
                     int skip_idx, float* __restrict__ out, int M)
{
    __shared__ float Wl[16][130];
    const int lane = threadIdx.x & 31;
    const int wave = threadIdx.x >> 5;
    const int half = lane >> 4;
    const int nrow = lane & 15;
    const int m0   = blockIdx.x * 128 + wave * 16;
    const bool active = (m0 < M);
    const float* Trow = T + (size_t)(m0 + nrow) * OUT_DIM;
    const float alpha = 1.0f / (1.0f + __expf(-skip[skip_idx]));

    v2f af[32];
    #pragma unroll
    for (int i = 0; i < 32; ++i) {
        if (active) af[i] = *(const v2f*)(Trow + 4 * i + 2 * half);
        else { af[i][0] = 0.0f; af[i][1] = 0.0f; }
    }

    for (int nt = 0; nt < 8; ++nt) {
        const int n0 = nt * 16;
        __syncthreads();
        for (int idx = threadIdx.x; idx < 128 * 16; idx += 256) {
            int k = idx >> 4, n = idx & 15;
            Wl[n][k] = W[k * OUT_DIM + n0 + n];
        }
        __syncthreads();
        if (!active) continue;

        const float bval = bias[n0 + nrow];
        v8f acc;
        #pragma unroll
        for (int j = 0; j < 8; ++j) acc[j] = bval;

        #pragma unroll
        for (int i = 0; i < 32; ++i) {
            v2f b;
            b[0] = Wl[nrow][4 * i + 2 * half];
            b[1] = Wl[nrow][4 * i + 2 * half + 1];
            acc = __builtin_amdgcn_wmma_f32_16x16x4_f32(
                false, af[i], false, b, (short)0, acc, false, false);
        }
        #pragma unroll
        for (int j = 0; j < 8; ++j) {
            const int row = m0 + j + 8 * half;
            const float f = feat[(size_t)row * OUT_DIM + n0 + nrow];
            const bool has = denom[(size_t)row * NH] > 0.0f;
            const float tr = has ? acc[j] : f;
            out[(size_t)row * OUT_DIM + n0 + nrow] = tr * alpha + f * (1.0f - alpha);
        }
    }
}

// ---------------- edge pass 1: attention logits + segment max ----------------
// one thread per (edge, head); 64B-aligned float4 gathers
__global__ void edge_att_kernel(const float* __restrict__ Ksrc, const float* __restrict__ Qdst,
                                const float* __restrict__ ra,   const float* __restrict__ pri,
                                const int* __restrict__ src,    const int* __restrict__ dst,
                                float* __restrict__ att, float* __restrict__ amax, int E)
{
    int tid = blockIdx.x * blockDim.x + threadIdx.x;
    if (tid >= E * NH) return;
    const int e = tid >> 3, h = tid & 7;
    const int s = src[e], d = dst[e];
    float kv[DKH], qv[DKH];
    load16(Ksrc + (size_t)s * OUT_DIM + h * DKH, kv);
    load16(Qdst + (size_t)d * OUT_DIM + h * DKH, qv);
    const float* rp = ra + h * DKH * DKH;

    float key[DKH];
    #pragma unroll
    for (int f = 0; f < DKH; ++f) key[f] = 0.0f;
    #pragma unroll 4
    for (int dd = 0; dd < DKH; ++dd) {
        const float kd = kv[dd];
        #pragma unroll
        for (int f = 0; f < DKH; ++f) key[f] = fmaf(kd, rp[dd * DKH + f], key[f]);
    }
    float a = 0.0f;
    #pragma unroll
    for (int f = 0; f < DKH; ++f) a = fmaf(key[f], qv[f], a);
    a *= pri[h] * 0.25f;   // 1/sqrt(16)

    att[(size_t)e * NH + h] = a;
    atomicMaxF(&amax[(size_t)d * NH + h], a);
}

// ---------------- edge pass 2: exp/denom + weighted message scatter ----------------
__global__ void edge_accum_kernel(const float* __restrict__ Vsrc, const float* __restrict__ rm,
                                  const int* __restrict__ src,    const int* __restrict__ dst,
                                  const float* __restrict__ att,  const float* __restrict__ amax,
                                  float* __restrict__ denom, float* __restrict__ traw, int E)
{
    int tid = blockIdx.x * blockDim.x + threadIdx.x;
    if (tid >= E * NH) return;
    const int e = tid >> 3, h = tid & 7;
    const int s = src[e], d = dst[e];
    const float ex = __expf(att[(size_t)e * NH + h] - amax[(size_t)d * NH + h]);
    atomicAdd(&denom[(size_t)d * NH + h], ex);

    float vv[DKH];
    load16(Vsrc + (size_t)s * OUT_DIM + h * DKH, vv);
    const float* rp = rm + h * DKH * DKH;
    float val[DKH];
    #pragma unroll
    for (int f = 0; f < DKH; ++f) val[f] = 0.0f;
    #pragma unroll 4
    for (int dd = 0; dd < DKH; ++dd) {
        const float vd = vv[dd];
        #pragma unroll
        for (int f = 0; f < DKH; ++f) val[f] = fmaf(vd, rp[dd * DKH + f], val[f]);
    }
    float* tp = traw + (size_t)d * OUT_DIM + h * DKH;
    #pragma unroll
    for (int f = 0; f < DKH; ++f) atomicAdd(&tp[f], ex * val[f]);
}

// ---------------- normalize messages: t = t_raw / max(denom, 1e-9) ----------------
__global__ void norm_kernel(float* __restrict__ traw, const float* __restrict__ denom, int n)
{
    int i = blockIdx.x * blockDim.x + threadIdx.x;
    if (i >= n) return;
    const int node = i >> 7;          // /128
    const int h = (i >> 4) & 7;       // head
    const float dn = denom[(size_t)node * NH + h];
    traw[i] = traw[i] / fmaxf(dn, 1e-9f);
}

// ---------------- host orchestration ----------------
extern "C" void kernel_launch(void* const* d_in, const int* in_sizes, int n_in,
                              void* d_out, int out_size, void* d_ws, size_t ws_size,
                              hipStream_t stream)
{
    const float* feat_a  = (const float*)d_in[0];
    const float* feat_b  = (const float*)d_in[1];
    const float* Wk      = (const float*)d_in[2];
    const float* bk      = (const float*)d_in[3];
    const float* Wq      = (const float*)d_in[4];
    const float* bq      = (const float*)d_in[5];
    const float* Wv      = (const float*)d_in[6];
    const float* bv      = (const float*)d_in[7];
    const float* Wa      = (const float*)d_in[8];
    const float* ba      = (const float*)d_in[9];
    const float* rel_att = (const float*)d_in[10];
    const float* rel_msg = (const float*)d_in[11];
    const float* rel_pri = (const float*)d_in[12];
    const float* skip    = (const float*)d_in[13];
    const int*   src0    = (const int*)d_in[14];
    const int*   dst0    = (const int*)d_in[15];
    const int*   src1    = (const int*)d_in[16];
    const int*   dst1    = (const int*)d_in[17];
    float* out = (float*)d_out;       // h_a [NA*128] then h_b [NB*128]
    float* ws  = (float*)d_ws;

    // workspace layout (floats)
    const size_t NFA = (size_t)NA * OUT_DIM;
    const size_t NFB = (size_t)NB * OUT_DIM;
    float* k_a = ws;
    float* v_a = k_a + NFA;
    float* q_a = v_a + NFA;
    float* k_b = q_a + NFA;
    float* v_b = k_b + NFB;
    float* q_b = v_b + NFB;
    float* att0 = q_b + NFB;                       // E*H
    float* att1 = att0 + (size_t)NEDGE * NH;
    float* amax_a = att1 + (size_t)NEDGE * NH;     // NA*H
    float* amax_b = amax_a + (size_t)NA * NH;
    float* denom_a = amax_b + (size_t)NB * NH;
    float* denom_b = denom_a + (size_t)NA * NH;
    float* traw_a = denom_b + (size_t)NB * NH;
    float* traw_b = traw_a + NFA;

    const int THREADS = 256;
    // 1) init: amax = -inf ; denom & t_raw = 0 (contiguous regions)
    {
        int n_amax = (NA + NB) * NH;
        fill_kernel<<<(n_amax + THREADS - 1) / THREADS, THREADS, 0, stream>>>(
            amax_a, -INFINITY, n_amax);
        int n_zero = (int)((NA + NB) * NH + NFA + NFB);   // denoms + traws (contiguous)
        fill_kernel<<<(n_zero + THREADS - 1) / THREADS, THREADS, 0, stream>>>(
            denom_a, 0.0f, n_zero);
    }

    // 2) fused K/Q/V projections (WMMA fp32), one pass over feat per node type
    const int GBLK = (NA + 127) / 128;   // 313 (NA==NB)
    gemm_kqv_kernel<<<GBLK, THREADS, 0, stream>>>(
        feat_a, Wk, Wq, Wv, bk, bq, bv, k_a, q_a, v_a, NA);
    gemm_kqv_kernel<<<GBLK, THREADS, 0, stream>>>(
        feat_b, Wk + 128 * 128, Wq + 128 * 128, Wv + 128 * 128,
        bk + 128, bq + 128, bv + 128, k_b, q_b, v_b, NB);

    // 3) edge attention logits + segment max
    const int EBLK = (NEDGE * NH + THREADS - 1) / THREADS;
    edge_att_kernel<<<EBLK, THREADS, 0, stream>>>(
        k_a, q_b, rel_att,            rel_pri,      src0, dst0, att0, amax_b, NEDGE);
    edge_att_kernel<<<EBLK, THREADS, 0, stream>>>(
        k_b, q_a, rel_att + NH * 256, rel_pri + NH, src1, dst1, att1, amax_a, NEDGE);

    // 4) softmax accumulate + message scatter
    edge_accum_kernel<<<EBLK, THREADS, 0, stream>>>(
        v_a, rel_msg,            src0, dst0, att0, amax_b, denom_b, traw_b, NEDGE);
    edge_accum_kernel<<<EBLK, THREADS, 0, stream>>>(
        v_b, rel_msg + NH * 256, src1, dst1, att1, amax_a, denom_a, traw_a, NEDGE);

    // 5) normalize messages in place
    const int NNORM = (int)NFA;
    norm_kernel<<<(NNORM + THREADS - 1) / THREADS, THREADS, 0, stream>>>(traw_a, denom_a, NNORM);
    norm_kernel<<<(NNORM + THREADS - 1) / THREADS, THREADS, 0, stream>>>(traw_b, denom_b, NNORM);

    // 6) output GEMM with fused skip-gate epilogue
    gemm_out_kernel<<<GBLK, THREADS, 0, stream>>>(
        traw_a, Wa,             ba,       feat_a, denom_a, skip, 0, out,       NA);
    gemm_out_kernel<<<GBLK, THREADS, 0, stream>>>(
        traw_b, Wa + 128 * 128, ba + 128, feat_b, denom_b, skip, 1, out + NFA, NB);
}